// NCA_55164559950510
// MI455X (gfx1250) — compile-verified
//
#include <hip/hip_runtime.h>
#include <cstdint>
#include <cstddef>

// Vegetation PDE step: memory-bound 5-point stencil + pointwise chemistry.
// MI455X strategy:
//   - stage w/b tiles (with replicate-pad halo) into LDS via CDNA5 async
//     global->LDS DMA (global_load_async_to_lds_b128/_b32, ASYNCcnt path)
//   - compute from LDS (ds_load), stream T in and both outputs out as b128
//   - single-use traffic (T, outputs) carries non-temporal cache hints so the
//     halo-reused w/b lines stay resident in the 192 MB L2.

#define IMG_H 1024
#define IMG_W 1024
#define NBATCH 16
#define TILE_W 128
#define TILE_H 32
#define PAD_ROWS (TILE_H + 2)      // 34 rows (halo top/bottom)
#define LDS_STRIDE 136             // floats/row; interior starts at col 4 -> 16B aligned
#define BLOCK_THREADS 256

// Real vector type (clang ext_vector) so nontemporal builtins accept it and
// loads/stores lower to b128.
typedef __attribute__((ext_vector_type(4))) float v4f;

__global__ __launch_bounds__(BLOCK_THREADS) void veg_step_kernel(
    const float* __restrict__ wIn,
    const float* __restrict__ bIn,
    const float* __restrict__ tIn,
    float* __restrict__ outW,
    float* __restrict__ outB)
{
    __shared__ float ldsW[PAD_ROWS * LDS_STRIDE];
    __shared__ float ldsB[PAD_ROWS * LDS_STRIDE];

    const int tid    = threadIdx.x;
    const int tileX0 = blockIdx.x * TILE_W;
    const int tileY0 = blockIdx.y * TILE_H;
    const int batch  = blockIdx.z;
    const size_t imgBase = (size_t)batch * IMG_H * IMG_W;

    // LDS byte offsets of the two staging tiles (low 32 bits of flat LDS addr).
    const uint32_t ldsWBase = (uint32_t)(uintptr_t)(&ldsW[0]);
    const uint32_t ldsBBase = (uint32_t)(uintptr_t)(&ldsB[0]);

    // ---- Async interior staging: B128 (4 floats / lane / issue), per field ----
    // 34 padded rows x 32 float4-quads = 1088 issues per field per block.
    {
        const float*   srcs[2]  = { wIn, bIn };
        const uint32_t bases[2] = { ldsWBase, ldsBBase };
#pragma unroll
        for (int f = 0; f < 2; ++f) {
            const float*   fld  = srcs[f];
            const uint32_t base = bases[f];
            for (int idx = tid; idx < PAD_ROWS * 32; idx += BLOCK_THREADS) {
                const int row = idx >> 5;      // padded row 0..33
                const int q   = idx & 31;      // float4 index within 128-wide interior
                int gy = tileY0 + row - 1;
                gy = gy < 0 ? 0 : (gy > IMG_H - 1 ? IMG_H - 1 : gy);  // replicate pad
                const float* src = fld + imgBase + (size_t)gy * IMG_W + tileX0 + q * 4;
                const uint32_t dst = base + (uint32_t)((row * LDS_STRIDE + 4 + q * 4) * 4);
                const unsigned long long ga = (unsigned long long)(uintptr_t)src;
                asm volatile("global_load_async_to_lds_b128 %0, %1, off"
                             :: "v"(dst), "v"(ga) : "memory");
            }
            // ---- Halo columns: B32 (left col -> lds col 3, right -> col 132) ----
            for (int idx = tid; idx < PAD_ROWS * 2; idx += BLOCK_THREADS) {
                const int side = (idx >= PAD_ROWS) ? 1 : 0;
                const int row  = side ? idx - PAD_ROWS : idx;
                int gy = tileY0 + row - 1;
                gy = gy < 0 ? 0 : (gy > IMG_H - 1 ? IMG_H - 1 : gy);
                int gx = side ? (tileX0 + TILE_W) : (tileX0 - 1);
                gx = gx < 0 ? 0 : (gx > IMG_W - 1 ? IMG_W - 1 : gx);  // replicate pad
                const float* src = fld + imgBase + (size_t)gy * IMG_W + gx;
                const int ldsCol = side ? (4 + TILE_W) : 3;
                const uint32_t dst = base + (uint32_t)((row * LDS_STRIDE + ldsCol) * 4);
                const unsigned long long ga = (unsigned long long)(uintptr_t)src;
                asm volatile("global_load_async_to_lds_b32 %0, %1, off"
                             :: "v"(dst), "v"(ga) : "memory");
            }
        }
    }

    // Drain this wave's async DMA, then make LDS visible across all 8 waves.
    asm volatile("s_wait_asynccnt 0x0" ::: "memory");
    __syncthreads();

    // ---- Compute: each thread owns a 4x4 patch; v4f in/out on globals ----
    const int tx = tid & 31;         // 32 col-groups of 4
    const int ty = tid >> 5;         // 8 row-groups of 4
    const int c0 = tx * 4;           // interior col base 0..124
    const int r0 = ty * 4;           // interior row base 0..28

    for (int rr = 0; rr < 4; ++rr) {
        const int r  = r0 + rr;
        const int lr = r + 1;                         // padded row
        const size_t gRow = imgBase + (size_t)(tileY0 + r) * IMG_W + tileX0 + c0;

        // T is read exactly once across the whole kernel: non-temporal load.
        const v4f t4 = __builtin_nontemporal_load((const v4f*)(tIn + gRow));

        v4f wo, bo;
#pragma unroll
        for (int cc = 0; cc < 4; ++cc) {
            const int lc = 4 + c0 + cc;               // padded col
            const float wC = ldsW[lr * LDS_STRIDE + lc];
            const float bC = ldsB[lr * LDS_STRIDE + lc];
            const float lapW = ldsW[(lr - 1) * LDS_STRIDE + lc]
                             + ldsW[(lr + 1) * LDS_STRIDE + lc]
                             + ldsW[lr * LDS_STRIDE + lc - 1]
                             + ldsW[lr * LDS_STRIDE + lc + 1]
                             - 4.0f * wC;
            const float lapB = ldsB[(lr - 1) * LDS_STRIDE + lc]
                             + ldsB[(lr + 1) * LDS_STRIDE + lc]
                             + ldsB[lr * LDS_STRIDE + lc - 1]
                             + ldsB[lr * LDS_STRIDE + lc + 1]
                             - 4.0f * bC;
            // uptake = UPTAKE_RATE * w * b * (1 + ETA * b^Q), UPTAKE=1, ETA=3, Q=1
            const float uptake = wC * bC * (1.0f + 3.0f * bC);
            // dw = (D2*lapW - SEEP*w + PRECIP - uptake) * TS
            wo[cc] = wC + (lapW - 0.2f * wC + 1.845f - uptake) * 1.0e-4f;
            // db = (D3*lapB - MORT*T*b + EFF*uptake) * TS
            bo[cc] = bC + (0.05f * lapB - 0.3f * t4[cc] * bC + 0.449f * uptake) * 1.0e-4f;
        }
        // Outputs are write-once, never re-read: non-temporal streaming stores.
        __builtin_nontemporal_store(wo, (v4f*)(outW + gRow));
        __builtin_nontemporal_store(bo, (v4f*)(outB + gRow));
    }
}

extern "C" void kernel_launch(void* const* d_in, const int* in_sizes, int n_in,
                              void* d_out, int out_size, void* d_ws, size_t ws_size,
                              hipStream_t stream) {
    (void)in_sizes; (void)n_in; (void)out_size; (void)d_ws; (void)ws_size;
    const float* w = (const float*)d_in[0];   // "w"
    const float* b = (const float*)d_in[1];   // "b"
    const float* T = (const float*)d_in[2];   // "T"
    float* out = (float*)d_out;
    const size_t N = (size_t)NBATCH * IMG_H * IMG_W;

    dim3 grid(IMG_W / TILE_W, IMG_H / TILE_H, NBATCH);  // 8 x 32 x 16
    dim3 block(BLOCK_THREADS);
    veg_step_kernel<<<grid, block, 0, stream>>>(w, b, T, out, out + N);
}